// DSS_Base_64364379898214
// MI455X (gfx1250) — compile-verified
//
#include <hip/hip_runtime.h>
#include <hip/hip_bf16.h>

// ---------------------------------------------------------------------------
// Bundle-recommendation propagation (LightGCN-style) for MI455X / gfx1250.
// Memory-bound SpMM pipeline in fp32; dense deg-PE GEMM done on the matrix
// unit with V_WMMA_F32_16X16X4_F32 (branch-free operand setup / writeback).
// ---------------------------------------------------------------------------

#define NU_ 200000
#define NI_ 100000
#define NB_ 50000
#define DD  64

static_assert(NI_ % 16 == 0, "enrich kernel assumes full 16-row WMMA tiles");

typedef __attribute__((ext_vector_type(2))) float v2f;
typedef __attribute__((ext_vector_type(8))) float v8f;

__device__ __forceinline__ void atomAddF(float* p, float v) {
    // lowers to global_atomic_add_f32 on gfx1250 (no CAS loop)
    unsafeAtomicAdd(p, v);
}

// ---- grid-stride zero ------------------------------------------------------
__global__ void k_zero(float* __restrict__ p, long long n) {
    long long i = (long long)blockIdx.x * blockDim.x + threadIdx.x;
    long long stride = (long long)gridDim.x * blockDim.x;
    for (; i < n; i += stride) p[i] = 0.0f;
}

// ---- grid-stride scale -----------------------------------------------------
__global__ void k_scale(float* __restrict__ p, long long n, float s) {
    long long i = (long long)blockIdx.x * blockDim.x + threadIdx.x;
    long long stride = (long long)gridDim.x * blockDim.x;
    for (; i < n; i += stride) p[i] *= s;
}

// ---- feat = concat(A,B); accA = A; accB = B (layer-0 term) -----------------
__global__ void k_init_prop(const float* __restrict__ A, const float* __restrict__ B,
                            float* __restrict__ feat,
                            float* __restrict__ accA, float* __restrict__ accB,
                            long long nAe, long long nTot) {
    long long i = (long long)blockIdx.x * blockDim.x + threadIdx.x;
    long long stride = (long long)gridDim.x * blockDim.x;
    for (; i < nTot; i += stride) {
        float x;
        if (i < nAe) { x = A[i]; accA[i] = x; }
        else         { x = B[i - nAe]; accB[i - nAe] = x; }
        feat[i] = x;
    }
}

// ---- SpMM scatter: one wave32 per edge, lanes cover features lane, lane+32 -
__global__ void k_spmm_edge(const int* __restrict__ row, const int* __restrict__ col,
                            const float* __restrict__ val,
                            const float* __restrict__ feat, float* __restrict__ out,
                            long long nE) {
    long long wid = ((long long)blockIdx.x * blockDim.x + threadIdx.x) >> 5;
    int lane = threadIdx.x & 31;
    if (wid >= nE) return;
    int r = row[wid];
    int c = col[wid];
    float v = val[wid];
    const float* src = feat + (long long)c * DD;
    float*       dst = out  + (long long)r * DD;
    // two fully-coalesced 128B gathers per wave + native fp32 atomics
    atomAddF(dst + lane,      v * src[lane]);
    atomAddF(dst + lane + 32, v * src[lane + 32]);
}

// ---- acc += l2norm(f) per node; one wave32 per 64-wide row -----------------
__global__ void k_l2norm_acc(const float* __restrict__ f,
                             float* __restrict__ accA, float* __restrict__ accB,
                             int nA, int n) {
    long long node = ((long long)blockIdx.x * blockDim.x + threadIdx.x) >> 5;
    int lane = threadIdx.x & 31;
    if (node >= n) return;
    const float* p = f + node * (long long)DD;
    float v0 = p[lane];
    float v1 = p[lane + 32];
    float s = v0 * v0 + v1 * v1;
    #pragma unroll
    for (int off = 16; off; off >>= 1) s += __shfl_xor(s, off, 32);
    float inv = 1.0f / fmaxf(sqrtf(s), 1e-12f);
    float* acc = (node < nA) ? (accA + node * (long long)DD)
                             : (accB + (node - nA) * (long long)DD);
    acc[lane]      += v0 * inv;
    acc[lane + 32] += v1 * inv;
}

// ---- enriched = UI_i + 0.05 * (log1p(item_degree) @ W_deg^T) via WMMA ------
// One wave computes a full 16x64 tile: 4x V_WMMA_F32_16X16X4_F32 (K=2 zero-
// padded to 4).  Only the tile-level exit is divergent (wave-uniform), so
// EXEC is all-ones at the WMMA as the ISA requires.  Operand setup and
// writeback are branch-free (NI is a multiple of 16).
__global__ void k_enrich_wmma(const float* __restrict__ ui_i,
                              const float* __restrict__ item_degree,
                              const float* __restrict__ W_deg,
                              float* __restrict__ out, int ni) {
    int wid  = (int)(((long long)blockIdx.x * blockDim.x + threadIdx.x) >> 5);
    int lane = threadIdx.x & 31;
    int m0 = wid * 16;
    if (m0 >= ni) return;                 // wave-uniform exit, full tiles only

    int lm = lane & 15;
    float mask = (lane < 16) ? 1.0f : 0.0f;   // lanes 16-31 carry K=2/3 zeros

    // A (16x4 fp32): lanes 0-15 hold K=0 (x) / K=1 (y); lanes 16-31 masked 0.
    int ia = m0 + lm;                          // in-bounds for all lanes
    v2f a;
    a.x = mask * log1pf(item_degree[2 * ia + 0]);
    a.y = mask * log1pf(item_degree[2 * ia + 1]);

    int Mbase = (lane < 16) ? 0 : 8;

    #pragma unroll
    for (int nt = 0; nt < 4; ++nt) {
        // B (4x16 fp32): lanes 0-15 hold K=0 (x) / K=1 (y) for N = nt*16+lm.
        int n = nt * 16 + lm;                  // W_deg is (64,2) row-major
        v2f b;
        b.x = mask * W_deg[2 * n + 0];
        b.y = mask * W_deg[2 * n + 1];

        v8f c = {};
        v8f d = __builtin_amdgcn_wmma_f32_16x16x4_f32(
            /*neg_a=*/false, a, /*neg_b=*/false, b,
            /*c_mod=*/(short)0, c, /*reuse_a=*/false, /*reuse_b=*/false);

        // D layout: VGPR r -> row M = Mbase + r, col N = nt*16 + lm.
        long long base = (long long)(m0 + Mbase) * DD + n;
        #pragma unroll
        for (int r = 0; r < 8; ++r) {
            long long off = base + (long long)r * DD;
            out[off] = fmaf(0.05f, d[r], ui_i[off]);
        }
    }
}

// ---- bundle_emb += 0.05 * size_pe_table[clip(bundle_size,0,10)] ------------
__global__ void k_size_pe(float* __restrict__ be, const int* __restrict__ bsz,
                          const float* __restrict__ table, int nb) {
    long long idx = (long long)blockIdx.x * blockDim.x + threadIdx.x;
    long long n = (long long)nb * DD;
    long long stride = (long long)gridDim.x * blockDim.x;
    for (; idx < n; idx += stride) {
        int b = (int)(idx >> 6);
        int d = (int)(idx & 63);
        int s = bsz[b];
        s = (s < 0) ? 0 : (s > 10 ? 10 : s);
        be[idx] += 0.05f * table[s * DD + d];
    }
}

// ---------------------------------------------------------------------------

static void run_propagate(const int* row, const int* col, const float* val, long long nE,
                          const float* A, int nA, const float* B, int nB,
                          float* outA, float* outB,
                          float* f0, float* f1, hipStream_t stream) {
    const int T = 256;
    int n = nA + nB;
    long long nTot = (long long)n * DD;
    long long nAe  = (long long)nA * DD;

    k_init_prop<<<4096, T, 0, stream>>>(A, B, f0, outA, outB, nAe, nTot);

    for (int layer = 0; layer < 2; ++layer) {
        k_zero<<<4096, T, 0, stream>>>(f1, nTot);
        long long nThreads = nE * 32;
        unsigned blocks = (unsigned)((nThreads + T - 1) / T);
        k_spmm_edge<<<blocks, T, 0, stream>>>(row, col, val, f0, f1, nE);
        long long nodeThreads = (long long)n * 32;
        unsigned nblocks = (unsigned)((nodeThreads + T - 1) / T);
        k_l2norm_acc<<<nblocks, T, 0, stream>>>(f1, outA, outB, nA, n);
        float* t = f0; f0 = f1; f1 = t;
    }
    const float inv3 = 1.0f / 3.0f;
    k_scale<<<2048, T, 0, stream>>>(outA, nAe, inv3);
    k_scale<<<2048, T, 0, stream>>>(outB, (long long)nB * DD, inv3);
}

extern "C" void kernel_launch(void* const* d_in, const int* in_sizes, int n_in,
                              void* d_out, int out_size, void* d_ws, size_t ws_size,
                              hipStream_t stream) {
    const float* users       = (const float*)d_in[0];
    const float* items       = (const float*)d_in[1];
    const float* bundles     = (const float*)d_in[2];
    const float* W_deg       = (const float*)d_in[3];
    const float* size_pe     = (const float*)d_in[4];
    const float* item_degree = (const float*)d_in[5];
    const int*   bundle_size = (const int*)d_in[6];
    const int*   ui_row = (const int*)d_in[7];
    const int*   ui_col = (const int*)d_in[8];
    const float* ui_val = (const float*)d_in[9];
    const int*   bi_row = (const int*)d_in[10];
    const int*   bi_col = (const int*)d_in[11];
    const float* bi_val = (const float*)d_in[12];
    const int*   ub_row = (const int*)d_in[13];
    const int*   ub_col = (const int*)d_in[14];
    const float* ub_val = (const float*)d_in[15];
    const int*   agg_row = (const int*)d_in[16];
    const int*   agg_col = (const int*)d_in[17];
    const float* agg_val = (const float*)d_in[18];

    long long nUIe  = in_sizes[7];    // 4,000,000
    long long nBIe  = in_sizes[10];   // 2,000,000
    long long nUBe  = in_sizes[13];   // 2,000,000
    long long nAGGe = in_sizes[16];   //   500,000

    // Output slices, concatenated in reference return order.
    float* out  = (float*)d_out;
    float* UI_u = out;
    float* UB_u = UI_u + (long long)NU_ * DD;
    float* BI_b = UB_u + (long long)NU_ * DD;
    float* BI_i = BI_b + (long long)NB_ * DD;
    float* UB_b = BI_i + (long long)NI_ * DD;
    float* UI_i = UB_b + (long long)NB_ * DD;
    float* BEmb = UI_i + (long long)NI_ * DD;

    // Workspace: two ping-pong feature buffers sized for the largest graph
    // (UI: NU+NI = 300000 nodes x 64 fp32).
    const long long MAXN = (long long)(NU_ + NI_) * DD;
    float* f0 = (float*)d_ws;
    float* f1 = f0 + MAXN;

    // Three propagations, sequential on the stream.
    run_propagate(ui_row, ui_col, ui_val, nUIe, users,   NU_, items,   NI_,
                  UI_u, UI_i, f0, f1, stream);
    run_propagate(bi_row, bi_col, bi_val, nBIe, bundles, NB_, items,   NI_,
                  BI_b, BI_i, f0, f1, stream);
    run_propagate(ub_row, ub_col, ub_val, nUBe, users,   NU_, bundles, NB_,
                  UB_u, UB_b, f0, f1, stream);

    const int T = 256;

    // enriched_UI_i = UI_i + 0.05 * deg_pe   (WMMA f32 16x16x4, K padded)
    {
        long long waves = (NI_ + 15) / 16;              // 6250 waves
        long long threads = waves * 32;
        unsigned blocks = (unsigned)((threads + T - 1) / T);
        k_enrich_wmma<<<blocks, T, 0, stream>>>(UI_i, item_degree, W_deg, f0, NI_);
    }

    // mean_emb = agg_graph @ enriched  (scatter into bundle_emb slice)
    k_zero<<<2048, T, 0, stream>>>(BEmb, (long long)NB_ * DD);
    {
        long long threads = nAGGe * 32;
        unsigned blocks = (unsigned)((threads + T - 1) / T);
        k_spmm_edge<<<blocks, T, 0, stream>>>(agg_row, agg_col, agg_val, f0, BEmb, nAGGe);
    }

    // bundle_emb = mean_emb + 0.05 * size_pe_table[clip(size)]
    k_size_pe<<<2048, T, 0, stream>>>(BEmb, bundle_size, size_pe, NB_);
}